// XSimGCL_31568009626128
// MI455X (gfx1250) — compile-verified
//
#include <hip/hip_runtime.h>

#define USER_NUM 100000
#define ITEM_NUM 100000
#define NTOT     (USER_NUM + ITEM_NUM)
#define DIM      64
#define N_EDGES  3200000
#define N_LAYERS 3
#define EPS_F    0.2f

typedef __attribute__((ext_vector_type(2))) float v2f;
typedef __attribute__((ext_vector_type(8))) float v8f;

__device__ __forceinline__ float signf(float x) {
    return (x > 0.0f) ? 1.0f : ((x < 0.0f) ? -1.0f : 0.0f);
}

// Guaranteed hardware no-return f32 atomics (never a CAS loop):
// two adds to [addr] and [addr+4] sharing one 64-bit address VGPR pair.
__device__ __forceinline__ void atomic_add2_f32_hw(float* addr, float vx, float vy) {
    asm volatile("global_atomic_add_f32 %0, %1, off\n\t"
                 "global_atomic_add_f32 %0, %2, off offset:4"
                 :
                 : "v"(addr), "v"(vx), "v"(vy)
                 : "memory");
}

// ---------------------------------------------------------------- zero fill
__global__ void xsim_zero_kernel(float* __restrict__ p, int n4) {
    int i      = blockIdx.x * blockDim.x + threadIdx.x;
    int stride = gridDim.x * blockDim.x;
    float4* p4 = (float4*)p;
    float4  z  = make_float4(0.f, 0.f, 0.f, 0.f);
    for (; i < n4; i += stride) p4[i] = z;
}

// ------------------------------------------------------- edge scatter (SpMM)
// One wave32 per edge: 32 lanes x float2 = one full 64-float row (256B).
// srcA serves rows [0, USER_NUM), srcB rows [USER_NUM, NTOT).
__global__ void xsim_scatter_kernel(const float* __restrict__ srcA,
                                    const float* __restrict__ srcB,
                                    float* __restrict__ outp,
                                    const int* __restrict__ asrc,
                                    const int* __restrict__ adst,
                                    const float* __restrict__ avals) {
    const int lane   = threadIdx.x & 31;
    const int wave   = (int)((blockIdx.x * blockDim.x + threadIdx.x) >> 5);
    const int nwaves = (int)((gridDim.x * blockDim.x) >> 5);

    for (int e = wave; e < N_EDGES; e += nwaves) {
        const int   s = asrc[e];
        const int   d = adst[e];
        const float v = avals[e];

        const float* row = (s < USER_NUM)
                               ? (srcA + (size_t)s * DIM)
                               : (srcB + (size_t)(s - USER_NUM) * DIM);

        // Prefetch next iteration's gather row (random access -> hide latency)
        const int e2 = e + nwaves;
        if (e2 < N_EDGES) {
            const int s2 = asrc[e2];
            const float* row2 = (s2 < USER_NUM)
                                    ? (srcA + (size_t)s2 * DIM)
                                    : (srcB + (size_t)(s2 - USER_NUM) * DIM);
            __builtin_prefetch(row2 + 2 * lane, 0, 0);
        }

        v2f x = *(const v2f*)(row + 2 * lane);
        float* orow = outp + (size_t)d * DIM + 2 * lane;
        atomic_add2_f32_hw(orow, x.x * v, x.y * v);
    }
}

// ------------------------------------------- noise injection + accumulation
// One wave32 per 16-row tile. Squared row norms of the noise tile are the
// diagonal of X * X^T, computed with 16 chained V_WMMA_F32_16X16X4_F32.
// f32 16x4 A-layout == 4x16 B-layout per lane, so the same registers serve
// as both operands and D accumulates the Gram matrix.
__global__ void xsim_post_kernel(float* __restrict__ ego,
                                 const float* __restrict__ noise_k,
                                 float* __restrict__ acc,
                                 float* __restrict__ cl) {
    const int lane = threadIdx.x & 31;
    const int half = lane >> 4;     // which K pair of the 16x4 A slice
    const int r    = lane & 15;     // row within tile owned by this lane
    const int wave = (int)((blockIdx.x * blockDim.x + threadIdx.x) >> 5);
    const int ntiles = NTOT / 16;   // 12500
    if (wave >= ntiles) return;     // wave-uniform exit keeps EXEC full for WMMA

    const size_t rowbase = (size_t)wave * 16;
    const float* nrow = noise_k + (rowbase + r) * DIM;

    float g; // squared L2 norm of noise row (rowbase + r)

#if __has_builtin(__builtin_amdgcn_wmma_f32_16x16x4_f32)
    v8f c = {};
#pragma unroll
    for (int ch = 0; ch < 16; ++ch) {
        v2f a;
        a.x = nrow[4 * ch + 2 * half];
        a.y = nrow[4 * ch + 2 * half + 1];
        // D(16x16) += A(16x4) x B(4x16); A == B == noise chunk -> Gram matrix
        c = __builtin_amdgcn_wmma_f32_16x16x4_f32(
                false, a, false, a, (short)0, c, false, false);
    }
    // diag[m]: m<8 -> c[m] on lane m ; m>=8 -> c[m-8] on lane m+16
    const int q = r & 7;
    float cand = c[0];
    cand = (q == 1) ? c[1] : cand;
    cand = (q == 2) ? c[2] : cand;
    cand = (q == 3) ? c[3] : cand;
    cand = (q == 4) ? c[4] : cand;
    cand = (q == 5) ? c[5] : cand;
    cand = (q == 6) ? c[6] : cand;
    cand = (q == 7) ? c[7] : cand;
    const int srcLane = (r < 8) ? r : (r + 16);
    g = __shfl(cand, srcLane, 32);
#else
    g = 0.0f;
#pragma unroll
    for (int ch = 0; ch < 16; ++ch) {
        float x0 = nrow[4 * ch + 2 * half];
        float x1 = nrow[4 * ch + 2 * half + 1];
        g += x0 * x0 + x1 * x1;
    }
    g += __shfl_xor(g, 16, 32); // combine the two half-lane partials
#endif

    const float inv = 1.0f / fmaxf(sqrtf(g), 1e-12f);

    // Coalesced elementwise pass: row by row, 32 lanes x float2 = 256B/row.
#pragma unroll
    for (int rr = 0; rr < 16; ++rr) {
        const float invr = __shfl(inv, rr, 32) * EPS_F;
        const size_t off = (rowbase + rr) * DIM + 2 * lane;

        v2f e  = *(v2f*)(ego + off);
        v2f nz = *(const v2f*)(noise_k + off);

        v2f o;
        o.x = e.x + signf(e.x) * nz.x * invr;
        o.y = e.y + signf(e.y) * nz.y * invr;
        *(v2f*)(ego + off) = o;

        v2f a2 = *(v2f*)(acc + off);
        a2.x += o.x * (1.0f / (float)N_LAYERS);
        a2.y += o.y * (1.0f / (float)N_LAYERS);
        *(v2f*)(acc + off) = a2;

        if (cl) { *(v2f*)(cl + off) = o; }
    }
}

// --------------------------------------------------------------------- host
extern "C" void kernel_launch(void* const* d_in, const int* in_sizes, int n_in,
                              void* d_out, int out_size, void* d_ws, size_t ws_size,
                              hipStream_t stream) {
    const float* user_emb = (const float*)d_in[0];
    const float* item_emb = (const float*)d_in[1];
    const int*   adj_src  = (const int*)d_in[2];
    const int*   adj_dst  = (const int*)d_in[3];
    const float* adj_vals = (const float*)d_in[4];
    const float* noise    = (const float*)d_in[5];

    float* out = (float*)d_out;
    float* acc = out;                          // [N*D] -> final (already /3)
    float* cl  = out + (size_t)NTOT * DIM;     // [N*D] -> layer-0 output

    float* buf0 = (float*)d_ws;                // ping-pong ego buffers
    float* buf1 = buf0 + (size_t)NTOT * DIM;

    const int ND  = NTOT * DIM;                // 12.8M floats
    const int ND4 = ND / 4;

    // acc region of d_out is poisoned -> zero it once
    xsim_zero_kernel<<<2048, 256, 0, stream>>>(acc, ND4);

    const float* curA = user_emb;  // layer 0 gathers straight from inputs
    const float* curB = item_emb;
    float* bufs[2] = {buf0, buf1};

    const int ntiles      = NTOT / 16;                  // 12500 waves
    const int post_blocks = (ntiles + 7) / 8;           // 8 waves/block

    for (int k = 0; k < N_LAYERS; ++k) {
        float* nxt = bufs[k & 1];
        xsim_zero_kernel<<<2048, 256, 0, stream>>>(nxt, ND4);
        xsim_scatter_kernel<<<4096, 256, 0, stream>>>(curA, curB, nxt,
                                                      adj_src, adj_dst, adj_vals);
        xsim_post_kernel<<<post_blocks, 256, 0, stream>>>(
            nxt, noise + (size_t)k * ND, acc, (k == 0) ? cl : nullptr);
        curA = nxt;
        curB = nxt + (size_t)USER_NUM * DIM;   // so (s-U)*D + U*D == s*D
    }
}